// NeighborhoodAttentionModule_6923487282486
// MI455X (gfx1250) — compile-verified
//
#include <hip/hip_runtime.h>

typedef __attribute__((ext_vector_type(2))) float v2f;
typedef __attribute__((ext_vector_type(8))) float v8f;

#define D_MODEL 256
#define A_DIM   64
#define K_NB    16
#define TILE    16
#define ATT_SCALE 0.125f   /* 64^-0.5 */
#define LN_EPS  1e-5f

static __device__ __forceinline__ v8f wmma_f32(v2f a, v2f b, v8f c) {
  return __builtin_amdgcn_wmma_f32_16x16x4_f32(false, a, false, b, (short)0, c, false, false);
}

__global__ __launch_bounds__(256)
void neigh_attn_kernel(const float* __restrict__ all_embs,
                       const int*   __restrict__ center_idx,
                       const int*   __restrict__ nb_idx,
                       const float* __restrict__ nb_weights,
                       const float* __restrict__ Wq,
                       const float* __restrict__ Wk,
                       const float* __restrict__ Wg,
                       const float* __restrict__ bg,
                       const float* __restrict__ Wo,
                       const float* __restrict__ bo,
                       const float* __restrict__ gamma,
                       const float* __restrict__ beta,
                       float*       __restrict__ out)
{
  __shared__ float sC[TILE][D_MODEL];     // center rows                (16 KB)
  __shared__ float sQ[TILE][A_DIM];       // Q projection               ( 4 KB)
  __shared__ float sBuf[TILE][D_MODEL];   // V -> gated ctx -> x        (16 KB)
  __shared__ float sGate[TILE][D_MODEL];  // sigmoid gate               (16 KB)
  __shared__ float sAttn[TILE][K_NB];
  __shared__ int   sNbIdx[TILE][K_NB];
  __shared__ float sNbW[TILE][K_NB];

  const int tid  = threadIdx.x;
  const int wave = tid >> 5;
  const int lane = tid & 31;
  const int lm   = lane & 15;   // M row (A frag) or N col (B/D frag) within 16x16 tile
  const int half = lane >> 4;   // selects K pair (A/B) or M half (C/D)
  const int b0   = blockIdx.x * TILE;

  // ---- Phase A: stage 16 center rows + neighbor indices/weights -------------
  {
    const int s = tid >> 4;           // 0..15 tile row
    const int p = tid & 15;
    const float* rp = all_embs + (size_t)center_idx[b0 + s] * D_MODEL;
    #pragma unroll
    for (int j = 0; j < 4; ++j) {
      const int c = (p + 16 * j) * 4;
      *(float4*)&sC[s][c] = *(const float4*)(rp + c);
    }
    sNbIdx[s][p] = nb_idx[(b0 + s) * K_NB + p];
    sNbW[s][p]   = nb_weights[(b0 + s) * K_NB + p];
  }
  __syncthreads();

  // ---- Phase B: Q = C @ Wq^T (waves 0-3, 1 tile) and
  //               gate = sigm(C @ Wg^T + bg) (16 tiles split 2/wave across all)
  if (wave < 4) {
    const int n = wave * 16 + lm;
    v8f acc = {};
    for (int k0 = 0; k0 < D_MODEL; k0 += 4) {
      const int kk = k0 + half * 2;
      v2f a; a.x = sC[lm][kk];            a.y = sC[lm][kk + 1];
      v2f b; b.x = Wq[n * D_MODEL + kk];  b.y = Wq[n * D_MODEL + kk + 1];
      acc = wmma_f32(a, b, acc);
    }
    #pragma unroll
    for (int r = 0; r < 8; ++r) sQ[half * 8 + r][n] = acc[r];
  }
  {
    // gate tiles: waves 0-3 -> tiles 8+2w, 9+2w ; waves 4-7 -> tiles 2(w-4), 2(w-4)+1
    const int base = (wave < 4) ? (8 + wave * 2) : ((wave - 4) * 2);
    for (int t = 0; t < 2; ++t) {
      const int n = (base + t) * 16 + lm;
      v8f acc = {};
      for (int k0 = 0; k0 < D_MODEL; k0 += 4) {
        const int kk = k0 + half * 2;
        v2f a; a.x = sC[lm][kk];            a.y = sC[lm][kk + 1];
        v2f b; b.x = Wg[n * D_MODEL + kk];  b.y = Wg[n * D_MODEL + kk + 1];
        acc = wmma_f32(a, b, acc);
      }
      const float bgn = bg[n];
      #pragma unroll
      for (int r = 0; r < 8; ++r)
        sGate[half * 8 + r][n] = __fdividef(1.0f, 1.0f + __expf(-(acc[r] + bgn)));
    }
  }
  __syncthreads();

  // ---- Phase C: V = Q @ Wk  (M=16, N=256, K=64); 2 N-tiles per wave --------
  for (int t = 0; t < 2; ++t) {
    const int n = (wave * 2 + t) * 16 + lm;
    v8f acc = {};
    for (int k0 = 0; k0 < A_DIM; k0 += 4) {
      const int kk = k0 + half * 2;
      v2f a; a.x = sQ[lm][kk];              a.y = sQ[lm][kk + 1];
      v2f b; b.x = Wk[kk * D_MODEL + n];    b.y = Wk[(kk + 1) * D_MODEL + n];
      acc = wmma_f32(a, b, acc);
    }
    #pragma unroll
    for (int r = 0; r < 8; ++r) sBuf[half * 8 + r][n] = acc[r];
  }
  __syncthreads();

  // ---- Phase D: logits[k] = (nbs[k] . V) * SCALE + log(clip(w)); softmax ---
  for (int t = 0; t < 2; ++t) {
    const int s = wave * 2 + t;
    const float* rowp = all_embs + (size_t)sNbIdx[s][lm] * D_MODEL + half * 128;
    float dotv = 0.0f;
    for (int j = 0; j < 128; j += 4) {
      const float4 e = *(const float4*)(rowp + j);
      const int c = half * 128 + j;
      dotv += e.x * sBuf[s][c]     + e.y * sBuf[s][c + 1]
            + e.z * sBuf[s][c + 2] + e.w * sBuf[s][c + 3];
    }
    dotv += __shfl_xor(dotv, 16, 32);                 // fold halves
    float lg = dotv * ATT_SCALE + __logf(fmaxf(sNbW[s][lm], 1e-6f));
    float mx = lg;
    for (int m = 8; m >= 1; m >>= 1) mx = fmaxf(mx, __shfl_xor(mx, m, 16));
    const float e = __expf(lg - mx);
    float sum = e;
    for (int m = 8; m >= 1; m >>= 1) sum += __shfl_xor(sum, m, 16);
    if (half == 0) sAttn[s][lm] = __fdividef(e, sum);
  }
  __syncthreads();

  // ---- Phase F: ctx = gate * (attn @ nbs), into sBuf -----------------------
  {
    const int s  = tid >> 4;
    const int c0 = (tid & 15) * 16;
    float4 a0 = {0,0,0,0}, a1 = {0,0,0,0}, a2 = {0,0,0,0}, a3 = {0,0,0,0};
    for (int k = 0; k < K_NB; ++k) {
      const float w = sAttn[s][k];
      const float* rp = all_embs + (size_t)sNbIdx[s][k] * D_MODEL + c0;
      const float4 e0 = *(const float4*)(rp + 0);
      const float4 e1 = *(const float4*)(rp + 4);
      const float4 e2 = *(const float4*)(rp + 8);
      const float4 e3 = *(const float4*)(rp + 12);
      a0.x += w*e0.x; a0.y += w*e0.y; a0.z += w*e0.z; a0.w += w*e0.w;
      a1.x += w*e1.x; a1.y += w*e1.y; a1.z += w*e1.z; a1.w += w*e1.w;
      a2.x += w*e2.x; a2.y += w*e2.y; a2.z += w*e2.z; a2.w += w*e2.w;
      a3.x += w*e3.x; a3.y += w*e3.y; a3.z += w*e3.z; a3.w += w*e3.w;
    }
    const float* g = &sGate[s][c0];
    float* d = &sBuf[s][c0];
    d[0]=a0.x*g[0];  d[1]=a0.y*g[1];  d[2]=a0.z*g[2];  d[3]=a0.w*g[3];
    d[4]=a1.x*g[4];  d[5]=a1.y*g[5];  d[6]=a1.z*g[6];  d[7]=a1.w*g[7];
    d[8]=a2.x*g[8];  d[9]=a2.y*g[9];  d[10]=a2.z*g[10];d[11]=a2.w*g[11];
    d[12]=a3.x*g[12];d[13]=a3.y*g[13];d[14]=a3.z*g[14];d[15]=a3.w*g[15];
  }
  __syncthreads();

  // ---- Phase G: enriched = [C, ctx] @ Wo^T  (M=16, N=256, K=512) -----------
  v8f accE[2];
  for (int t = 0; t < 2; ++t) {
    const int n = (wave * 2 + t) * 16 + lm;
    v8f acc = {};
    for (int k0 = 0; k0 < 2 * D_MODEL; k0 += 4) {
      const int kk = k0 + half * 2;
      v2f a;
      if (k0 < D_MODEL) { a.x = sC[lm][kk];             a.y = sC[lm][kk + 1]; }
      else              { a.x = sBuf[lm][kk - D_MODEL]; a.y = sBuf[lm][kk - D_MODEL + 1]; }
      v2f b; b.x = Wo[n * 2 * D_MODEL + kk]; b.y = Wo[n * 2 * D_MODEL + kk + 1];
      acc = wmma_f32(a, b, acc);
    }
    accE[t] = acc;
  }
  __syncthreads();  // all A-frag reads of sBuf(ctx) done

  // ---- Phase H: x = enriched + bo + center, into sBuf ----------------------
  for (int t = 0; t < 2; ++t) {
    const int n = (wave * 2 + t) * 16 + lm;
    const float bon = bo[n];
    #pragma unroll
    for (int r = 0; r < 8; ++r) {
      const int row = half * 8 + r;
      sBuf[row][n] = accE[t][r] + bon + sC[row][n];
    }
  }
  __syncthreads();

  // ---- Phase I: LayerNorm per row, write output ----------------------------
  for (int t = 0; t < 2; ++t) {
    const int row = wave * 2 + t;
    const int c0  = lane * 8;
    const float4 x0 = *(const float4*)&sBuf[row][c0];
    const float4 x1 = *(const float4*)&sBuf[row][c0 + 4];
    float sum = x0.x + x0.y + x0.z + x0.w + x1.x + x1.y + x1.z + x1.w;
    float sq  = x0.x*x0.x + x0.y*x0.y + x0.z*x0.z + x0.w*x0.w
              + x1.x*x1.x + x1.y*x1.y + x1.z*x1.z + x1.w*x1.w;
    for (int m = 16; m >= 1; m >>= 1) {
      sum += __shfl_xor(sum, m, 32);
      sq  += __shfl_xor(sq,  m, 32);
    }
    const float mu  = sum * (1.0f / 256.0f);
    const float var = sq  * (1.0f / 256.0f) - mu * mu;
    const float rs  = rsqrtf(var + LN_EPS);
    const float4 g0 = *(const float4*)(gamma + c0);
    const float4 g1 = *(const float4*)(gamma + c0 + 4);
    const float4 be0 = *(const float4*)(beta + c0);
    const float4 be1 = *(const float4*)(beta + c0 + 4);
    float4 o0, o1;
    o0.x = (x0.x - mu) * rs * g0.x + be0.x;
    o0.y = (x0.y - mu) * rs * g0.y + be0.y;
    o0.z = (x0.z - mu) * rs * g0.z + be0.z;
    o0.w = (x0.w - mu) * rs * g0.w + be0.w;
    o1.x = (x1.x - mu) * rs * g1.x + be1.x;
    o1.y = (x1.y - mu) * rs * g1.y + be1.y;
    o1.z = (x1.z - mu) * rs * g1.z + be1.z;
    o1.w = (x1.w - mu) * rs * g1.w + be1.w;
    float* op = out + (size_t)(b0 + row) * D_MODEL + c0;
    *(float4*)op       = o0;
    *(float4*)(op + 4) = o1;
  }
}

extern "C" void kernel_launch(void* const* d_in, const int* in_sizes, int n_in,
                              void* d_out, int out_size, void* d_ws, size_t ws_size,
                              hipStream_t stream) {
  const float* all_embs   = (const float*)d_in[0];
  const int*   center_idx = (const int*)  d_in[1];
  const int*   nb_idx     = (const int*)  d_in[2];
  const float* nb_w       = (const float*)d_in[3];
  const float* Wq         = (const float*)d_in[4];
  const float* Wk         = (const float*)d_in[5];
  const float* Wg         = (const float*)d_in[6];
  const float* bg         = (const float*)d_in[7];
  const float* Wo         = (const float*)d_in[8];
  const float* bo         = (const float*)d_in[9];
  const float* gamma      = (const float*)d_in[10];
  const float* beta       = (const float*)d_in[11];
  float* out = (float*)d_out;

  const int B = in_sizes[1];         // center_idx element count
  const int grid = B / TILE;         // B = 32768 -> 2048 blocks
  hipLaunchKernelGGL(neigh_attn_kernel, dim3(grid), dim3(256), 0, stream,
                     all_embs, center_idx, nb_idx, nb_w,
                     Wq, Wk, Wg, bg, Wo, bo, gamma, beta, out);
}